// DistMultDecoder_15264313770419
// MI455X (gfx1250) — compile-verified
//
#include <hip/hip_runtime.h>
#include <hip/hip_bf16.h>

typedef __attribute__((ext_vector_type(2))) float v2f;
typedef __attribute__((ext_vector_type(8))) float v8f;

#define DMODEL 256
#define WAVES_PER_BLOCK 8
#define ROWS_PER_TILE 16

// scores[n] = sigmoid( sum_d row[n,d] * rel[d] * col[n,d] )
// One wave32 computes a 16-row tile as diag( (row .* rel) x col^T ) using
// v_wmma_f32_16x16x4_f32, accumulating over K=256 in steps of 4.
__global__ __launch_bounds__(WAVES_PER_BLOCK * 32)
void distmult_wmma_kernel(const float* __restrict__ inputs_row,
                          const float* __restrict__ inputs_col,
                          const float* __restrict__ relations,
                          const int* __restrict__ relation_index,
                          float* __restrict__ out,
                          int n_rows, int n_tiles) {
    __shared__ float rel_lds[DMODEL];
    __shared__ float tile_lds[WAVES_PER_BLOCK * ROWS_PER_TILE * ROWS_PER_TILE];

    const int tid  = threadIdx.x;
    const int lane = tid & 31;
    const int wv   = tid >> 5;

    // Stage the selected relation vector (256 floats) into LDS once per block.
    {
        const int ridx = relation_index[0];
        rel_lds[tid] = relations[(size_t)ridx * DMODEL + tid];
    }
    __syncthreads();

    const int tile = blockIdx.x * WAVES_PER_BLOCK + wv;
    if (tile >= n_tiles) return;   // whole-wave early out: EXEC stays all-ones below

    const int n0 = tile * ROWS_PER_TILE;

    // WMMA f32 16x16x4 operand layout (wave32):
    //   A (16x4): lane&15 = M,  lane>>4 selects K in {0,1} vs {2,3}; 2 VGPRs = K pair
    //   B (4x16): lane&15 = N,  lane>>4 selects K in {0,1} vs {2,3}; 2 VGPRs = K pair
    // With A = (row .* rel) tile and B[k][j] = col[n0+j][k], both operands for a
    // lane come from the SAME address pattern: row n0+(lane&15), cols k+2*(lane>>4).
    const int m  = lane & 15;
    const int ko = (lane >> 4) << 1;            // 0 or 2
    int nr = n0 + m;
    if (nr >= n_rows) nr = n_rows - 1;          // clamp (keeps EXEC uniform)

    const float* rp = inputs_row + (size_t)nr * DMODEL + ko;
    const float* cp = inputs_col + (size_t)nr * DMODEL + ko;

    v8f acc = {};

    #pragma unroll 8
    for (int k = 0; k < DMODEL; k += 4) {
        if ((k & 31) == 0) {
            // one prefetch per 128B line, 512B ahead (global_prefetch_b8)
            __builtin_prefetch(rp + k + 128, 0, 1);
            __builtin_prefetch(cp + k + 128, 0, 1);
        }
        const float2 rv   = *reinterpret_cast<const float2*>(rp + k);     // global_load_b64
        const float2 cv   = *reinterpret_cast<const float2*>(cp + k);     // global_load_b64
        const float2 relv = *reinterpret_cast<const float2*>(&rel_lds[k + ko]); // ds_load_b64

        v2f a; a.x = rv.x * relv.x; a.y = rv.y * relv.y;   // A = row .* rel
        v2f b; b.x = cv.x;          b.y = cv.y;            // B = col^T tile

        acc = __builtin_amdgcn_wmma_f32_16x16x4_f32(
            /*neg_a=*/false, a, /*neg_b=*/false, b,
            /*c_mod=*/(short)0, acc, /*reuse_a=*/false, /*reuse_b=*/false);
    }

    // Stage this wave's 16x16 result tile to LDS (DS ops are in-order per wave),
    // then lanes 0..15 pull the diagonal and store coalesced.
    float* tbuf = &tile_lds[wv * ROWS_PER_TILE * ROWS_PER_TILE];
    const int mrow = (lane >> 4) << 3;          // 0 or 8: M offset of this lane half
    #pragma unroll
    for (int r = 0; r < 8; ++r) {
        tbuf[(mrow + r) * ROWS_PER_TILE + m] = acc[r];   // element (M=mrow+r, N=m)
    }

    if (lane < ROWS_PER_TILE) {
        const int n = n0 + lane;
        if (n < n_rows) {
            const float s = tbuf[lane * (ROWS_PER_TILE + 1)];   // diagonal (lane,lane)
            out[n] = 1.0f / (1.0f + __expf(-s));
        }
    }
}

extern "C" void kernel_launch(void* const* d_in, const int* in_sizes, int n_in,
                              void* d_out, int out_size, void* d_ws, size_t ws_size,
                              hipStream_t stream) {
    const float* inputs_row = (const float*)d_in[0];
    const float* inputs_col = (const float*)d_in[1];
    const float* relations  = (const float*)d_in[2];
    const int*   rel_idx    = (const int*)d_in[3];
    float*       out        = (float*)d_out;

    const int n_rows  = in_sizes[0] / DMODEL;                  // 500000
    const int n_tiles = (n_rows + ROWS_PER_TILE - 1) / ROWS_PER_TILE;   // 31250
    const int blocks  = (n_tiles + WAVES_PER_BLOCK - 1) / WAVES_PER_BLOCK;

    distmult_wmma_kernel<<<blocks, WAVES_PER_BLOCK * 32, 0, stream>>>(
        inputs_row, inputs_col, relations, rel_idx, out, n_rows, n_tiles);
}